// Compress70_45320494907496
// MI455X (gfx1250) — compile-verified
//
#include <hip/hip_runtime.h>
#include <stdint.h>

// ---------------------------------------------------------------------------
// CDNA5 async global->LDS copy (gfx1250). The low 32 bits of a flat pointer
// into __shared__ memory are the raw LDS byte offset (ISA 10.2 aperture rule),
// which is exactly what the VDST operand of global_load_async_to_lds wants.
// Tracked with ASYNCcnt; we wait with s_wait_asynccnt before the barrier.
// ---------------------------------------------------------------------------
__device__ __forceinline__ void async_copy_b64_to_lds(void* lds_dst, const void* gsrc) {
    uint32_t lds_off = (uint32_t)(uintptr_t)lds_dst;
    asm volatile("global_load_async_to_lds_b64 %0, %1, off"
                 :: "v"(lds_off), "v"(gsrc)
                 : "memory");
}

__device__ __forceinline__ void wait_async_lds() {
    asm volatile("s_wait_asynccnt 0" ::: "memory");
}

// ---------------------------------------------------------------------------
// T_lv2: [2,128,128,128]  (C2=128, H=W=128), k=3, s=1, p=1
// One block handles one (b, p) output row for 32 channels.
// ---------------------------------------------------------------------------
__global__ __launch_bounds__(256)
void fold_lv2_kernel(const float* __restrict__ lv2,
                     const int*   __restrict__ hidx_g,
                     float*       __restrict__ out2)
{
    const int p   = blockIdx.x;      // output row 0..127
    const int b   = blockIdx.y;      // batch 0..1
    const int cz  = blockIdx.z;      // channel group (32 channels each)
    const int tid = threadIdx.x;

    __shared__ __align__(16) int sIdx[3][128];   // h_index rows p-1..p+1

    // Async-stage 3 index rows (2 ints per b64 transfer -> 192 transfers).
    if (tid < 3 * 64) {
        const int r   = tid >> 6;          // 0..2  -> row p-1+r
        const int pr  = tid & 63;          // int-pair within row
        const int row = p - 1 + r;
        if (row >= 0 && row < 128) {
            async_copy_b64_to_lds(&sIdx[r][pr * 2],
                                  hidx_g + (size_t)b * 16384 + (size_t)row * 128 + pr * 2);
        }
        // invalid rows: LDS left stale, but never read (j-bounds check below)
    }
    wait_async_lds();
    __syncthreads();

    // Per-thread: fixed spatial column q. Tap offsets + validity are
    // channel-invariant: clamp invalid taps to offset 0 with weight 0 so the
    // channel loop is 9 unconditional loads + 9 FMAs (no branches).
    const int q     = tid & 127;
    const int clane = tid >> 7;            // 0..1
    int   off[9];
    float w[9];
#pragma unroll
    for (int k = 0; k < 9; ++k) {
        const int dy = k / 3 - 1;
        const int dx = k % 3 - 1;
        const int jy = p - dy;
        const int jx = q - dx;
        int o = -1;
        if (jy >= 0 && jy < 128 && jx >= 0 && jx < 128) {
            const int i  = sIdx[1 - dy][jx];          // LDS row (jy)-(p-1) = 1-dy
            const int sy = (i >> 7) + dy;
            const int sx = (i & 127) + dx;
            if (sy >= 0 && sy < 128 && sx >= 0 && sx < 128) o = sy * 128 + sx;
        }
        w[k]   = (o >= 0) ? (1.0f / 9.0f) : 0.0f;
        off[k] = (o >= 0) ? o : 0;
    }

    const int c0 = cz * 32 + clane;
#pragma unroll 1
    for (int ci = 0; ci < 16; ++ci) {
        const int c = c0 + ci * 2;
        const float* __restrict__ src = lv2 + ((size_t)(b * 128 + c)) * 16384;
        float acc = 0.0f;
#pragma unroll
        for (int k = 0; k < 9; ++k) {
            acc = fmaf(w[k], src[off[k]], acc);
        }
        __builtin_nontemporal_store(
            acc, &out2[(((size_t)b * 128 + c) * 128 + p) * 128 + q]);
    }
}

// ---------------------------------------------------------------------------
// T_lv1: [2,64,256,256]  (C1=64, 2H=2W=256), k=6, s=2, p=2
// One block handles one (b, P) output row for 16 channels.
// ---------------------------------------------------------------------------
__global__ __launch_bounds__(256)
void fold_lv1_kernel(const float* __restrict__ lv1,
                     const int*   __restrict__ hidx_g,
                     float*       __restrict__ out1)
{
    const int P   = blockIdx.x;      // output row 0..255
    const int b   = blockIdx.y;
    const int cz  = blockIdx.z;      // channel group (16 channels each)
    const int tid = threadIdx.x;     // Q 0..255

    const int m  = P >> 1;
    const int py = P & 1;

    __shared__ __align__(16) int sIdx[3][128];   // h_index rows m-1..m+1

    if (tid < 3 * 64) {
        const int r   = tid >> 6;
        const int pr  = tid & 63;
        const int row = m - 1 + r;
        if (row >= 0 && row < 128) {
            async_copy_b64_to_lds(&sIdx[r][pr * 2],
                                  hidx_g + (size_t)b * 16384 + (size_t)row * 128 + pr * 2);
        }
    }
    wait_async_lds();
    __syncthreads();

    const int Q  = tid;
    const int n  = Q >> 1;
    const int px = Q & 1;

    int   off[9];
    float w[9];
#pragma unroll
    for (int k = 0; k < 9; ++k) {
        const int dy = k / 3 - 1;
        const int dx = k % 3 - 1;
        const int y  = m + dy;                    // patch location in 128-grid
        const int x  = n + dx;
        int o = -1;
        if (y >= 0 && y < 128 && x >= 0 && x < 128) {
            const int i  = sIdx[dy + 1][x];       // LDS row (y)-(m-1) = dy+1
            const int sy = 2 * (i >> 7)  + py - 2 * dy;
            const int sx = 2 * (i & 127) + px - 2 * dx;
            if (sy >= 0 && sy < 256 && sx >= 0 && sx < 256) o = sy * 256 + sx;
        }
        w[k]   = (o >= 0) ? (1.0f / 9.0f) : 0.0f;
        off[k] = (o >= 0) ? o : 0;
    }

    const int c0 = cz * 16;
#pragma unroll 1
    for (int ci = 0; ci < 16; ++ci) {
        const int c = c0 + ci;
        const float* __restrict__ src = lv1 + ((size_t)(b * 64 + c)) * 65536;
        float acc = 0.0f;
#pragma unroll
        for (int k = 0; k < 9; ++k) {
            acc = fmaf(w[k], src[off[k]], acc);
        }
        __builtin_nontemporal_store(
            acc, &out1[(((size_t)b * 64 + c) * 256 + P) * 256 + Q]);
    }
}

// ---------------------------------------------------------------------------
// Launch: d_in = {ref_lv1, ref_lv2, h_index}; d_out = [T_lv2 | T_lv1] flat.
// ---------------------------------------------------------------------------
extern "C" void kernel_launch(void* const* d_in, const int* in_sizes, int n_in,
                              void* d_out, int out_size, void* d_ws, size_t ws_size,
                              hipStream_t stream) {
    (void)in_sizes; (void)n_in; (void)out_size; (void)d_ws; (void)ws_size;

    const float* lv1  = (const float*)d_in[0];   // [2,64,256,256]
    const float* lv2  = (const float*)d_in[1];   // [2,128,128,128]
    const int*   hidx = (const int*)d_in[2];     // [2,16384] (harness passes ints as int32)

    float* out2 = (float*)d_out;                          // 2*128*128*128 = 4194304
    float* out1 = (float*)d_out + (size_t)2 * 128 * 128 * 128;  // 2*64*256*256

    fold_lv2_kernel<<<dim3(128, 2, 4), 256, 0, stream>>>(lv2, hidx, out2);
    fold_lv1_kernel<<<dim3(256, 2, 4), 256, 0, stream>>>(lv1, hidx, out1);
}